// ScalarPooler_20100446945820
// MI455X (gfx1250) — compile-verified
//
#include <hip/hip_runtime.h>
#include <hip/hip_bf16.h>

typedef __attribute__((ext_vector_type(2))) float v2f;
typedef __attribute__((ext_vector_type(8))) float v8f;

#define NEG_INF_F (-1e30f)

// One wave32 per output row. Lane l owns subject positions l and l+32.
// Pooled vector computed with V_WMMA_F32_16X16X4_F32 (exact f32 math):
//   A[m][k] = softmax_weight[l0+k]  (broadcast over M)
//   B[k][n] = E[l0+k][n]            (embedding chunk, via LDS transpose)
//   D[m][n] = pooled[n]             (all rows equal; acc[0] -> lane n holds pooled[n])
__global__ __launch_bounds__(256) void attn_pool_dot_wmma_kernel(
    const float* __restrict__ subj_emb,   // [N_SUBJ, 16]
    const float* __restrict__ attn_w,     // [16]
    const float* __restrict__ attn_b,     // [1]
    const float* __restrict__ user_bias,  // [N_USERS]
    const float* __restrict__ item_bias,  // [N_ITEMS]
    const float* __restrict__ global_bias,// [1]
    const int*   __restrict__ user_idx,   // [B]
    const int*   __restrict__ item_idx,   // [B]
    const int*   __restrict__ fav,        // [B, L]
    const int*   __restrict__ book,       // [B, L]
    float*       __restrict__ out,        // [B]
    int nRows, int L)
{
    __shared__ __align__(16) float sE[8 * 52 * 16]; // per-wave 52x16 tile
    __shared__ float sW[8 * 52];                    // per-wave 52 weights

    const int lane = threadIdx.x & 31;
    const int wv   = threadIdx.x >> 5;
    const int row  = blockIdx.x * 8 + wv;
    const bool active = row < nRows;      // wave-uniform

    float* E = sE + wv * (52 * 16);
    float* W = sW + wv * 52;

    // Zero the whole per-wave tile once (pad rows/weights must be 0; also
    // guards against garbage being multiplied by zero weights).
    #pragma unroll
    for (int j = 0; j < 26; ++j) E[j * 32 + lane] = 0.0f;
    W[lane] = 0.0f;
    if (lane < 20) W[32 + lane] = 0.0f;

    // attention weights (uniform across wave -> scalarizable)
    const float4* wp = (const float4*)attn_w;
    const float4 w0 = wp[0], w1 = wp[1], w2 = wp[2], w3 = wp[3];
    const float  bscal = attn_b[0];

    float pooled[2];

    for (int pool = 0; pool < 2; ++pool) {
        const int* idxbase = (pool == 0) ? fav : book;
        float pd = 0.0f;

        if (active) {
            const int p0 = lane;
            const int p1 = lane + 32;
            const bool has0 = (p0 < L);
            const bool has1 = (p1 < L);

            const size_t rb = (size_t)row * (size_t)L;
            int i0 = has0 ? idxbase[rb + p0] : 0;
            int i1 = has1 ? idxbase[rb + p1] : 0;

            // gather embedding rows (64B each) — hot in L2
            const float4* r0p = (const float4*)(subj_emb + (size_t)i0 * 16);
            const float4* r1p = (const float4*)(subj_emb + (size_t)i1 * 16);
            float4 a0 = r0p[0], a1 = r0p[1], a2 = r0p[2], a3 = r0p[3];
            float4 b0 = r1p[0], b1 = r1p[1], b2 = r1p[2], b3 = r1p[3];

            float s0 = a0.x*w0.x + a0.y*w0.y + a0.z*w0.z + a0.w*w0.w
                     + a1.x*w1.x + a1.y*w1.y + a1.z*w1.z + a1.w*w1.w
                     + a2.x*w2.x + a2.y*w2.y + a2.z*w2.z + a2.w*w2.w
                     + a3.x*w3.x + a3.y*w3.y + a3.z*w3.z + a3.w*w3.w + bscal;
            float s1 = b0.x*w0.x + b0.y*w0.y + b0.z*w0.z + b0.w*w0.w
                     + b1.x*w1.x + b1.y*w1.y + b1.z*w1.z + b1.w*w1.w
                     + b2.x*w2.x + b2.y*w2.y + b2.z*w2.z + b2.w*w2.w
                     + b3.x*w3.x + b3.y*w3.y + b3.z*w3.z + b3.w*w3.w + bscal;

            bool v0 = has0 && (i0 != 0);
            bool v1 = has1 && (i1 != 0);

            // any(real) across the wave (shuffle-OR, wave32-safe)
            int anyv = (v0 || v1) ? 1 : 0;
            #pragma unroll
            for (int off = 16; off >= 1; off >>= 1)
                anyv |= __shfl_xor(anyv, off, 32);
            if (!anyv && p0 == 0) v0 = true;   // torch per-row fallback rule

            s0 = v0 ? s0 : NEG_INF_F;
            s1 = v1 ? s1 : NEG_INF_F;

            // softmax over <=64 lane-slots (masked at -1e30)
            float m = fmaxf(s0, s1);
            #pragma unroll
            for (int off = 16; off >= 1; off >>= 1)
                m = fmaxf(m, __shfl_xor(m, off, 32));
            float e0 = v0 ? __expf(s0 - m) : 0.0f;
            float e1 = v1 ? __expf(s1 - m) : 0.0f;
            float ssum = e0 + e1;
            #pragma unroll
            for (int off = 16; off >= 1; off >>= 1)
                ssum += __shfl_xor(ssum, off, 32);
            const float inv = 1.0f / ssum;

            // stage tile + weights to LDS (per-wave region)
            if (has0) {
                float4* ep = (float4*)(E + p0 * 16);
                ep[0] = a0; ep[1] = a1; ep[2] = a2; ep[3] = a3;
                W[p0] = e0 * inv;
            }
            if (has1) {
                float4* ep = (float4*)(E + p1 * 16);
                ep[0] = b0; ep[1] = b1; ep[2] = b2; ep[3] = b3;
                W[p1] = e1 * inv;
            }
        }
        __syncthreads();

        if (active) {
            // weighted pooling via f32 WMMA, K accumulated 4 at a time (52 = 13*4)
            v8f acc = {};
            const int koff = (lane < 16) ? 0 : 2;   // ISA 16x4 f32 A/B lane layout
            const int n = lane & 15;
            #pragma unroll
            for (int c = 0; c < 13; ++c) {
                v2f av, bv;
                av.x = W[4 * c + koff];
                av.y = W[4 * c + koff + 1];
                bv.x = E[(4 * c + koff) * 16 + n];
                bv.y = E[(4 * c + koff + 1) * 16 + n];
                acc = __builtin_amdgcn_wmma_f32_16x16x4_f32(
                    false, av, false, bv, (short)0, acc, false, false);
            }
            pd = acc[0];   // lane (l & 15) holds pooled[l & 15]
        }
        __syncthreads();
        pooled[pool] = pd;
    }

    if (active) {
        // dot over 16 dims: lane d holds u[d]*i[d]; reduce within 16-lane halves
        float du = pooled[0] * pooled[1];
        #pragma unroll
        for (int off = 8; off >= 1; off >>= 1)
            du += __shfl_xor(du, off, 32);
        if (lane == 0) {
            out[row] = du + user_bias[user_idx[row]]
                          + item_bias[item_idx[row]]
                          + global_bias[0];
        }
    }
}

extern "C" void kernel_launch(void* const* d_in, const int* in_sizes, int n_in,
                              void* d_out, int out_size, void* d_ws, size_t ws_size,
                              hipStream_t stream) {
    const float* subj_emb   = (const float*)d_in[0];
    const float* attn_w     = (const float*)d_in[1];
    const float* attn_b     = (const float*)d_in[2];
    const float* user_bias  = (const float*)d_in[3];
    const float* item_bias  = (const float*)d_in[4];
    const float* global_b   = (const float*)d_in[5];
    const int*   user_idx   = (const int*)d_in[6];
    const int*   item_idx   = (const int*)d_in[7];
    const int*   fav        = (const int*)d_in[8];
    const int*   book       = (const int*)d_in[9];
    float*       out        = (float*)d_out;

    const int B = in_sizes[6];
    const int L = in_sizes[8] / B;
    const int blocks = (B + 7) / 8;   // 8 row-waves per 256-thread block

    hipLaunchKernelGGL(attn_pool_dot_wmma_kernel, dim3(blocks), dim3(256), 0, stream,
                       subj_emb, attn_w, attn_b, user_bias, item_bias, global_b,
                       user_idx, item_idx, fav, book, out, B, L);
}